// PageManager_75445395521780
// MI455X (gfx1250) — compile-verified
//
#include <hip/hip_runtime.h>
#include <hip/hip_bf16.h>
#include <stdint.h>

#define NUM_PAGES   1048576
#define TPP         32
#define SLOTS       256
#define MPPS        4096
#define NCHUNK      128
#define CHUNK_INTS  (NUM_PAGES / NCHUNK)   /* 8192 ints per chunk */

/* workspace layout (int32 slots) */
#define WS_CHUNK_COUNT   0      /* 128: free pages per chunk            */
#define WS_CHUNK_PREFIX  128    /* 128: exclusive prefix of counts      */
#define WS_RANK          256    /* 256: allocation rank per slot        */
#define WS_NEEDS         512    /* 256: slot needs a new page?          */
#define WS_K             768    /* total #allocations requested         */
#define WS_NUMFREE       769    /* total free pages in [1,N)            */
#define WS_FREELIST      1024   /* 256: first K free page indices       */

/* ------------------------------------------------------------------ */
/* Kernel A: stream page_status -> out (int->float), fused free-count */
/* 128 blocks x 256 threads, 8 x int4 per thread = 8192 ints/block    */
/* ------------------------------------------------------------------ */
__global__ __launch_bounds__(256)
void ps_copy_count(const int* __restrict__ ps, float* __restrict__ out_ps,
                   int* __restrict__ ws) {
  __shared__ int red[8];
  const int tid   = threadIdx.x;
  const int blk   = blockIdx.x;
  const int base4 = blk * (CHUNK_INTS / 4);
  int cnt = 0;
#pragma unroll
  for (int it = 0; it < 8; ++it) {
    const int idx4 = base4 + it * 256 + tid;
    if (it + 1 < 8)  /* gfx1250 global_prefetch_b8 for the next tile */
      __builtin_prefetch(ps + (size_t)(idx4 + 256) * 4, 0, 1);
    const int4 v  = ((const int4*)ps)[idx4];
    const int  gi = idx4 * 4;
    cnt += (v.x == 0 && gi != 0);   /* index 0 is never a free page */
    cnt += (v.y == 0);
    cnt += (v.z == 0);
    cnt += (v.w == 0);
    ((float4*)out_ps)[idx4] =
        make_float4((float)v.x, (float)v.y, (float)v.z, (float)v.w);
  }
  /* wave32 reduce, then cross-wave via LDS */
#pragma unroll
  for (int off = 16; off > 0; off >>= 1) cnt += __shfl_down(cnt, off, 32);
  const int wave = tid >> 5, lane = tid & 31;
  if (lane == 0) red[wave] = cnt;
  __syncthreads();
  if (tid == 0) {
    int s = 0;
    for (int w = 0; w < 8; ++w) s += red[w];
    ws[WS_CHUNK_COUNT + blk] = s;
  }
}

/* ------------------------------------------------------------------ */
/* Kernel P: page_map copy through the CDNA5 async global->LDS mover  */
/* Double-buffered GLOBAL_LOAD_ASYNC_TO_LDS_B128 + s_wait_asynccnt    */
/* ------------------------------------------------------------------ */
__global__ __launch_bounds__(256)
void pm_copy_async(const int* __restrict__ pm, float* __restrict__ out_pm) {
  __shared__ int sh[2048];                 /* two 4 KB buffers */
  const int tid   = threadIdx.x;
  const int base4 = blockIdx.x * (CHUNK_INTS / 4);

  auto issue = [&](int it) {
    const int idx4 = base4 + it * 256 + tid;
    /* low 32 bits of the flat shared address == LDS byte offset */
    unsigned lds_off = (unsigned)(uintptr_t)(&sh[(it & 1) * 1024 + tid * 4]);
    uint64_t gaddr   = (uint64_t)(uintptr_t)(pm + (size_t)idx4 * 4);
    asm volatile("global_load_async_to_lds_b128 %0, %1, off"
                 :: "v"(lds_off), "v"(gaddr)
                 : "memory");
  };

  issue(0);
#pragma unroll
  for (int it = 0; it < 8; ++it) {
    if (it < 7) {
      issue(it + 1);                                  /* keep 2 in flight */
      asm volatile("s_wait_asynccnt 1" ::: "memory"); /* oldest done      */
    } else {
      asm volatile("s_wait_asynccnt 0" ::: "memory");
    }
    const int4 v = *(const int4*)(&sh[(it & 1) * 1024 + tid * 4]);
    const int idx4 = base4 + it * 256 + tid;
    ((float4*)out_pm)[idx4] =
        make_float4((float)v.x, (float)v.y, (float)v.z, (float)v.w);
  }
}

/* ------------------------------------------------------------------ */
/* Kernel B: per-slot scalar updates + rank prefix + chunk prefix      */
/* ------------------------------------------------------------------ */
__global__ __launch_bounds__(SLOTS)
void slot_setup(const int* __restrict__ seq_in, const int* __restrict__ npu_in,
                const int* __restrict__ cp_in,
                float* __restrict__ out_ns, float* __restrict__ out_nn,
                float* __restrict__ out_cp, float* __restrict__ out_ncpp,
                int* __restrict__ ws) {
  __shared__ int sneeds[SLOTS];
  const int i  = threadIdx.x;
  const int sq = seq_in[i];
  const int cp = cp_in[i];
  const int ns = sq + ((cp == -1) ? 0 : 1);
  const int nn = (ns + TPP - 1) / TPP;
  const int ncpp = (ns == 0) ? 0 : ((ns - 1) % TPP);
  const int needs = (nn > npu_in[i]) ? 1 : 0;
  out_ns[i]   = (float)ns;
  out_nn[i]   = (float)nn;
  out_cp[i]   = (float)cp;      /* overwritten by apply_alloc if needed */
  out_ncpp[i] = (float)ncpp;
  sneeds[i]        = needs;
  ws[WS_NEEDS + i] = needs;
  __syncthreads();
  if (i == 0) {                  /* tiny serial scans (256 + 128 adds) */
    int run = 0;
    for (int s = 0; s < SLOTS; ++s) { ws[WS_RANK + s] = run; run += sneeds[s]; }
    ws[WS_K] = run;
    int tot = 0;
    for (int c = 0; c < NCHUNK; ++c) {
      ws[WS_CHUNK_PREFIX + c] = tot;
      tot += ws[WS_CHUNK_COUNT + c];
    }
    ws[WS_NUMFREE] = tot;
  }
}

/* ------------------------------------------------------------------ */
/* Kernel C: ordered extraction of the first K free page indices.      */
/* One wave32 per chunk; ballot_w32 gives the in-order compaction.     */
/* Only chunks overlapping the first K free pages do any walking.      */
/* ------------------------------------------------------------------ */
__global__ __launch_bounds__(32)
void collect_free(const int* __restrict__ ps, int* __restrict__ ws) {
  const int c    = blockIdx.x;
  const int lane = threadIdx.x;
  const int K    = ws[WS_K];
  const int pre  = ws[WS_CHUNK_PREFIX + c];
  const int cnt  = ws[WS_CHUNK_COUNT + c];
  if (K == 0 || pre >= K || cnt == 0) return;
  int offset = pre;
  const int base = c * CHUNK_INTS;
  for (int it = 0; it < CHUNK_INTS / 32; ++it) {
    const int idx = base + it * 32 + lane;
    const bool freep = (idx != 0) && (ps[idx] == 0);
    const unsigned bal = __builtin_amdgcn_ballot_w32(freep);
    const int myp = __popc(bal & ((1u << lane) - 1u));
    if (freep) {
      const int r = offset + myp;
      if (r < K) ws[WS_FREELIST + r] = idx;
    }
    offset += __popc(bal);
    if (offset >= K) break;
  }
}

/* ------------------------------------------------------------------ */
/* Kernel D: apply the <=256 allocations onto the copied outputs.      */
/* k-th needing slot gets k-th free page; exhaustion -> page 0.        */
/* ------------------------------------------------------------------ */
__global__ __launch_bounds__(SLOTS)
void apply_alloc(const int* __restrict__ npu_in,
                 float* __restrict__ out_ps, float* __restrict__ out_pm,
                 float* __restrict__ out_cp, const int* __restrict__ ws) {
  const int i = threadIdx.x;
  if (ws[WS_NEEDS + i]) {
    const int k       = ws[WS_RANK + i];
    const int numfree = ws[WS_NUMFREE];
    const int nfp     = (k < numfree) ? ws[WS_FREELIST + k] : 0;
    out_ps[nfp] = 1.0f;
    out_pm[(size_t)i * MPPS + npu_in[i]] = (float)nfp;
    out_cp[i] = (float)nfp;
  }
}

extern "C" void kernel_launch(void* const* d_in, const int* in_sizes, int n_in,
                              void* d_out, int out_size, void* d_ws, size_t ws_size,
                              hipStream_t stream) {
  const int* ps  = (const int*)d_in[0];
  const int* pm  = (const int*)d_in[1];
  const int* seq = (const int*)d_in[2];
  const int* npu = (const int*)d_in[3];
  const int* cp  = (const int*)d_in[4];
  /* d_in[5] (current_page_position) is not needed: new_cpp derives from new_seq */

  float* out      = (float*)d_out;
  float* out_ps   = out;                       /* 1048576 */
  float* out_pm   = out + NUM_PAGES;           /* 1048576 */
  float* out_ns   = out + 2 * NUM_PAGES;       /* 256 */
  float* out_nn   = out_ns + SLOTS;            /* 256 */
  float* out_cp   = out_nn + SLOTS;            /* 256 */
  float* out_ncpp = out_cp + SLOTS;            /* 256 */
  int* ws = (int*)d_ws;

  hipLaunchKernelGGL(ps_copy_count, dim3(NCHUNK), dim3(256), 0, stream,
                     ps, out_ps, ws);
  hipLaunchKernelGGL(pm_copy_async, dim3(NCHUNK), dim3(256), 0, stream,
                     pm, out_pm);
  hipLaunchKernelGGL(slot_setup, dim3(1), dim3(SLOTS), 0, stream,
                     seq, npu, cp, out_ns, out_nn, out_cp, out_ncpp, ws);
  hipLaunchKernelGGL(collect_free, dim3(NCHUNK), dim3(32), 0, stream,
                     ps, ws);
  hipLaunchKernelGGL(apply_alloc, dim3(1), dim3(SLOTS), 0, stream,
                     npu, out_ps, out_pm, out_cp, ws);
}